// KalmanPredictor_27496380629556
// MI455X (gfx1250) — compile-verified
//
#include <hip/hip_runtime.h>
#include <math.h>

#define DT     0.1f
#define DEBAT  2.6f
#define TPB    256
#define HIST_T 20
#define ROWF   44   // padded floats per batch history row in LDS (40 + 4 pad, 16B aligned, gcd(44,64)=4 -> ~2-way bank conflicts max)

typedef int v4i __attribute__((ext_vector_type(4)));
typedef __attribute__((address_space(1))) v4i* gv4p;   // global (device) AS
typedef __attribute__((address_space(3))) v4i* lv4p;   // LDS (shared) AS
typedef __attribute__((address_space(3))) void* as3_vp;

__device__ __forceinline__ void async_copy_b128(const void* g, void* l) {
#if __has_builtin(__builtin_amdgcn_global_load_async_to_lds_b128)
  __builtin_amdgcn_global_load_async_to_lds_b128((gv4p)(v4i*)(void*)g, (lv4p)(v4i*)l, 0, 0);
#else
  unsigned lds_off = (unsigned)(unsigned long long)(as3_vp)l;
  unsigned long long ga = (unsigned long long)g;
  asm volatile("global_load_async_to_lds_b128 %0, %1, off"
               :: "v"(lds_off), "v"(ga) : "memory");
#endif
}

__device__ __forceinline__ void wait_async0() {
#if __has_builtin(__builtin_amdgcn_s_wait_asynccnt)
  __builtin_amdgcn_s_wait_asynccnt(0);
#else
  asm volatile("s_wait_asynccnt 0x0" ::: "memory");
#endif
}

__global__ __launch_bounds__(TPB) void ekf_kernel(
    const float* __restrict__ hist,
    const float* __restrict__ angle_std,
    const float* __restrict__ wheel_std,
    const float* __restrict__ velocity_std,
    const float* __restrict__ acceleration_std,
    const float* __restrict__ LR,
    const float* __restrict__ LQ,
    float* __restrict__ out,
    int B, int len_pred)
{
  __shared__ float sh[TPB * ROWF];   // 45056 B

  const int tid = threadIdx.x;
  const int b   = blockIdx.x * TPB + tid;

  // ---- stage this block's history tile into LDS via async copy (coalesced 16B chunks) ----
  const long long blk_e0 = (long long)blockIdx.x * TPB * (HIST_T * 2);
  const long long total  = (long long)B * (HIST_T * 2);
  const float* gbase = hist + blk_e0;

  if (blk_e0 + (long long)TPB * (HIST_T * 2) <= total) {
    // full block: no per-chunk guards (uniform scalar branch)
#pragma unroll
    for (int i = 0; i < HIST_T * 2 / 4; ++i) {    // 10 chunks per thread
      int c  = i * TPB + tid;                     // chunk id within block tile
      int t  = c / 10;                            // local batch row (chunks never cross rows: 40%4==0)
      int j0 = (c % 10) * 4;
      async_copy_b128(gbase + c * 4, &sh[t * ROWF + j0]);
    }
  } else {
#pragma unroll
    for (int i = 0; i < HIST_T * 2 / 4; ++i) {
      int c = i * TPB + tid;
      long long e = blk_e0 + (long long)c * 4;
      if (e + 3 < total) {
        int t  = c / 10;
        int j0 = (c % 10) * 4;
        async_copy_b128(gbase + c * 4, &sh[t * ROWF + j0]);
      }
    }
  }
  wait_async0();
  __syncthreads();

  if (b >= B) return;
  const float* zrow = &sh[tid * ROWF];

  // ---- batch-invariant constants (Q = LQ LQ^T is diagonal for this problem: off-diag exactly 0) ----
  float q[6];
#pragma unroll
  for (int i = 0; i < 6; ++i) {
    float s = 0.f;
#pragma unroll
    for (int k = 0; k < 6; ++k) { float v = LQ[i * 6 + k]; s += v * v; }
    q[i] = s;
  }
  const float lr00 = LR[0], lr01 = LR[1], lr10 = LR[2], lr11 = LR[3];
  const float r00 = lr00 * lr00 + lr01 * lr01;
  const float r01 = lr00 * lr10 + lr01 * lr11;
  const float r11 = lr10 * lr10 + lr11 * lr11;

  // ---- init state ----
  const float z0x = zrow[0], z0y = zrow[1], z1x = zrow[2], z1y = zrow[3];
  const float v0x = (z1x - z0x) / DT;
  const float v0y = (z1y - z0y) / DT;
  float x0 = z0x, x1 = z0y;
  float x2 = atan2f(v0y, v0x);
  float x3 = sqrtf(v0x * v0x + v0y * v0y);
  float x4 = 0.f, x5 = 0.f;

  float P[6][6];
#pragma unroll
  for (int i = 0; i < 6; ++i)
#pragma unroll
    for (int j = 0; j < 6; ++j) P[i][j] = 0.f;
  P[0][0] = r00; P[0][1] = r01; P[1][0] = r01; P[1][1] = r11;
  { const float a = angle_std[0];        P[2][2] = a * a; }
  { const float a = velocity_std[0];     P[3][3] = a * a; }
  { const float a = wheel_std[0];        P[4][4] = a * a; }
  { const float a = acceleration_std[0]; P[5][5] = a * a; }

  // ---- EKF predict: X update + in-place P = J P J^T + Q exploiting J = I + sparse(rows 2..5) ----
  auto predict = [&]() {
    const float acc = x5, w = x4;
    const float v_half = x3 + DT * acc * 0.5f;
    const float tan_w  = __tanf(w);
    const bool  mask   = fabsf(tan_w) > 1.0f;
    const float tan_nz = mask ? tan_w : 1.0f;
    const float dd     = DEBAT * DEBAT;
    const float curv_big = 1.0f / sqrtf(dd / (tan_nz * tan_nz) + 0.25f * dd);
    const float curv   = mask ? curv_big : (tan_w / DEBAT);
    const float ang_half = x2 + DT * v_half * curv * 0.5f;

    const float cos_a = __cosf(ang_half), sin_a = __sinf(ang_half);
    const float cos_w = __cosf(w);
    const float J32 = DT * curv;
    const float J31 = DT * (sin_a + v_half * cos_a * J32);
    const float J30 = DT * (cos_a - v_half * sin_a * J32);
    const float u   = 1.0f + 0.25f * tan_nz * tan_nz;
    const float u15 = u * sqrtf(u);
    const float cw2 = cos_w * cos_w;
    const float bigv = DT * v_half / (cw2 * 2.0f * u15 * DEBAT);
    const float smlv = DT * v_half / (cw2 * DEBAT);
    const float J42 = mask ? bigv : smlv;
    const float J41 =  DT * v_half * cos_a * J42;
    const float J40 = -DT * v_half * sin_a * J42;
    const float J52 = DT * DT * curv;
    const float J51 =  DT * v_half * cos_a * J52;
    const float J50 = -DT * v_half * sin_a * J52;
    const float J20 = -DT * v_half * sin_a;
    const float J21 =  DT * v_half * cos_a;

    const float c0 = __cosf(x2), s0 = __sinf(x2);     // uses pre-update angle
    const float nx0 = x0 + DT * x3 * c0;
    const float nx1 = x1 + DT * x3 * s0;
    const float nx2 = x2 + DT * x3 * curv;
    const float nx3 = x3 + DT * x5;
    x0 = nx0; x1 = nx1; x2 = nx2; x3 = nx3; x4 = w; x5 = acc;

    // T = J @ P  (row AXPYs, update order 5,4,3,2 keeps needed rows "old")
#pragma unroll
    for (int j = 0; j < 6; ++j) P[5][j] += J50 * P[0][j] + J51 * P[1][j] + J52 * P[2][j] + DT * P[3][j];
#pragma unroll
    for (int j = 0; j < 6; ++j) P[4][j] += J40 * P[0][j] + J41 * P[1][j] + J42 * P[2][j];
#pragma unroll
    for (int j = 0; j < 6; ++j) P[3][j] += J30 * P[0][j] + J31 * P[1][j] + J32 * P[2][j];
#pragma unroll
    for (int j = 0; j < 6; ++j) P[2][j] += J20 * P[0][j] + J21 * P[1][j];
    // P = T @ J^T  (col AXPYs, same ordering trick)
#pragma unroll
    for (int i = 0; i < 6; ++i) P[i][5] += J50 * P[i][0] + J51 * P[i][1] + J52 * P[i][2] + DT * P[i][3];
#pragma unroll
    for (int i = 0; i < 6; ++i) P[i][4] += J40 * P[i][0] + J41 * P[i][1] + J42 * P[i][2];
#pragma unroll
    for (int i = 0; i < 6; ++i) P[i][3] += J30 * P[i][0] + J31 * P[i][1] + J32 * P[i][2];
#pragma unroll
    for (int i = 0; i < 6; ++i) P[i][2] += J20 * P[i][0] + J21 * P[i][1];
    // + Q (diagonal)
#pragma unroll
    for (int i = 0; i < 6; ++i) P[i][i] += q[i];
  };

  // ---- 19 history predict+update steps ----
#pragma unroll 1
  for (int s = 1; s < HIST_T; ++s) {
    predict();
    const float zx = zrow[2 * s], zy = zrow[2 * s + 1];
    const float y0v = zx - x0, y1v = zy - x1;
    const float S00 = P[0][0] + r00, S01 = P[0][1] + r01;
    const float S10 = P[1][0] + r01, S11 = P[1][1] + r11;
    const float inv = 1.0f / (S00 * S11 - S01 * S10);
    const float i00 =  S11 * inv, i01 = -S01 * inv;
    const float i10 = -S10 * inv, i11 =  S00 * inv;
    float K0[6], K1[6];
#pragma unroll
    for (int i = 0; i < 6; ++i) {
      K0[i] = P[i][0] * i00 + P[i][1] * i10;
      K1[i] = P[i][0] * i01 + P[i][1] * i11;
    }
    x0 += K0[0] * y0v + K1[0] * y1v;
    x1 += K0[1] * y0v + K1[1] * y1v;
    x2 += K0[2] * y0v + K1[2] * y1v;
    x3 += K0[3] * y0v + K1[3] * y1v;
    x4 += K0[4] * y0v + K1[4] * y1v;
    x5 += K0[5] * y0v + K1[5] * y1v;
    x5 = fmodf(x5, 6.28318530717958647692f);
    float P0[6], P1[6];
#pragma unroll
    for (int j = 0; j < 6; ++j) { P0[j] = P[0][j]; P1[j] = P[1][j]; }
#pragma unroll
    for (int i = 0; i < 6; ++i)
#pragma unroll
      for (int j = 0; j < 6; ++j)
        P[i][j] -= K0[i] * P0[j] + K1[i] * P1[j];
  }

  // ---- len_pred prediction-only steps with output ----
  float* outb = out + (size_t)b * (size_t)len_pred * 5;
#pragma unroll 1
  for (int t = 0; t < len_pred; ++t) {
    predict();
    const float s0 = sqrtf(P[0][0]);
    const float s1 = sqrtf(P[1][1]);
    const float c  = P[0][1] / sqrtf(s0 * s1);
    float* o = outb + t * 5;
    o[0] = x0; o[1] = x1; o[2] = s0; o[3] = s1; o[4] = c;
  }
}

extern "C" void kernel_launch(void* const* d_in, const int* in_sizes, int n_in,
                              void* d_out, int out_size, void* d_ws, size_t ws_size,
                              hipStream_t stream) {
  (void)n_in; (void)d_ws; (void)ws_size;
  const float* hist             = (const float*)d_in[0];
  const float* angle_std        = (const float*)d_in[1];
  const float* wheel_std        = (const float*)d_in[2];
  const float* velocity_std     = (const float*)d_in[3];
  const float* acceleration_std = (const float*)d_in[4];
  const float* LR               = (const float*)d_in[5];
  const float* LQ               = (const float*)d_in[6];

  const int B        = in_sizes[0] / (HIST_T * 2);
  const int len_pred = out_size / (B * 5);
  const int grid     = (B + TPB - 1) / TPB;

  hipLaunchKernelGGL(ekf_kernel, dim3(grid), dim3(TPB), 0, stream,
                     hist, angle_std, wheel_std, velocity_std, acceleration_std,
                     LR, LQ, (float*)d_out, B, len_pred);
}